// PrefGP_65901978189913
// MI455X (gfx1250) — compile-verified
//
#include <hip/hip_runtime.h>
#include <math.h>

// ---------------- problem constants ----------------
constexpr int MN = 4096;   // number of points
constexpr int DD = 8;      // feature dim
constexpr int NS_ITERS = 36;   // Newton-Schulz iterations for inverse
constexpr int NEWTON_ITERS = 30;

typedef __attribute__((ext_vector_type(2))) float v2f;
typedef __attribute__((ext_vector_type(8))) float v8f;
typedef unsigned int u32x4 __attribute__((ext_vector_type(4)));
typedef unsigned int u32x8 __attribute__((ext_vector_type(8)));

// ---------------- fp32 WMMA GEMM:  C = alpha*(A@B) + beta*Dm ----------------
// Block tile 64(M) x 128(N), BK=32, double-buffered LDS.
// A tiles are staged by the Tensor Data Mover (tensor_load_to_lds) with
// hardware padding to a [64][36] layout; B tiles are lane-staged transposed.
// 8 waves; each wave computes a 32x32 tile as a 2x2 grid of 16x16 WMMA
// accumulators using V_WMMA_F32_16X16X4_F32.
#define BM 64
#define BN 128
#define BK 32
#define LDP 36   // padded row stride (floats); gcd(36,64)=4 -> conflict-free frags

__global__ __launch_bounds__(256)
void gemm_wmma_f32(float* __restrict__ C, const float* __restrict__ A,
                   const float* __restrict__ B, const float* __restrict__ Dm,
                   float alpha, float beta, int n) {
    __shared__ float ldsA[2][BM][LDP];
    __shared__ float ldsB[2][BN][LDP];   // B stored transposed: [n][k]

    const int tid     = threadIdx.x;
    const int lane    = tid & 31;
    const int wave    = tid >> 5;
    const int waveRow = wave >> 2;    // 0..1
    const int waveCol = wave & 3;     // 0..3
    const int half    = lane >> 4;    // 0: lanes 0-15, 1: lanes 16-31
    const int m16     = lane & 15;
    const int rowBase = blockIdx.y * BM;
    const int colBase = blockIdx.x * BN;
    const int nChunks = n / BK;

    // TDM descriptor group 1 (constant): 4B elements, pad enabled
    // (pad_interval code 4 -> every 32 DWORDs, pad_amount code 3 -> 4 DWORDs),
    // tensor = n x n row-major, tile = BK x BM.
    const unsigned g1w0 = (2u << 16) | (1u << 20) | (4u << 22) | (3u << 25);
    const unsigned g1w1 = ((unsigned)n & 0xFFFFu) << 16;                       // dim0 lo
    const unsigned g1w2 = ((unsigned)n >> 16) | (((unsigned)n & 0xFFFFu) << 16); // dim0 hi | dim1 lo
    const unsigned g1w3 = ((unsigned)n >> 16) | ((unsigned)BK << 16);          // dim1 hi | tile_dim0
    const unsigned g1w4 = (unsigned)BM;                                        // tile_dim1 | tile_dim2=0
    const unsigned g1w5 = (unsigned)n;                                         // dim0 stride lo
    const u32x8 g1 = { g1w0, g1w1, g1w2, g1w3, g1w4, g1w5, 0u, 0u };

    auto issue_tdm_A = [&](int chunk, int buf) {
        if (wave == 0) {
            unsigned long long ga =
                (unsigned long long)(uintptr_t)(A + (size_t)rowBase * n + (size_t)chunk * BK);
            unsigned ldsOff = (unsigned)(uintptr_t)&ldsA[buf][0][0];
            // group0: count=1, lds_addr, global_addr[56:0], type=2 (bits 127:126)
            u32x4 g0 = { 1u, ldsOff, (unsigned)ga,
                         (unsigned)(ga >> 32) | 0x80000000u };
            asm volatile("tensor_load_to_lds %0, %1" :: "s"(g0), "s"(g1) : "memory");
        }
    };

    v8f acc[2][2] = {};
    float rB[16];

    // ---- prologue: stage chunk 0 into buffer 0 ----
    issue_tdm_A(0, 0);
    #pragma unroll
    for (int i = 0; i < 16; ++i) {
        int flat = tid + i * 256;
        int kr = flat >> 7, c = flat & 127;
        rB[i] = B[(size_t)kr * n + colBase + c];
    }
    #pragma unroll
    for (int i = 0; i < 16; ++i) {
        int flat = tid + i * 256;
        int kr = flat >> 7, c = flat & 127;
        ldsB[0][c][kr] = rB[i];
    }
    __builtin_amdgcn_s_wait_tensorcnt(0);
    __syncthreads();

    int buf = 0;
    for (int ch = 0; ch < nChunks; ++ch) {
        const int  nxt     = ch + 1;
        const bool hasNext = nxt < nChunks;

        // prefetch next B tile into registers; kick off next A tile via TDM
        if (hasNext) {
            #pragma unroll
            for (int i = 0; i < 16; ++i) {
                int flat = tid + i * 256;
                int kr = flat >> 7, c = flat & 127;
                rB[i] = B[(size_t)(nxt * BK + kr) * n + colBase + c];
            }
            issue_tdm_A(nxt, buf ^ 1);
        }

        // ---- compute on current buffer ----
        #pragma unroll
        for (int kk = 0; kk < BK; kk += 4) {
            const int klo = kk + half * 2;  // even -> 8B aligned LDS reads
            v2f a0 = *(const v2f*)&ldsA[buf][waveRow * 32 +  0 + m16][klo];
            v2f a1 = *(const v2f*)&ldsA[buf][waveRow * 32 + 16 + m16][klo];
            v2f b0 = *(const v2f*)&ldsB[buf][waveCol * 32 +  0 + m16][klo];
            v2f b1 = *(const v2f*)&ldsB[buf][waveCol * 32 + 16 + m16][klo];
            acc[0][0] = __builtin_amdgcn_wmma_f32_16x16x4_f32(false, a0, false, b0, (short)0, acc[0][0], false, false);
            acc[0][1] = __builtin_amdgcn_wmma_f32_16x16x4_f32(false, a0, false, b1, (short)0, acc[0][1], false, false);
            acc[1][0] = __builtin_amdgcn_wmma_f32_16x16x4_f32(false, a1, false, b0, (short)0, acc[1][0], false, false);
            acc[1][1] = __builtin_amdgcn_wmma_f32_16x16x4_f32(false, a1, false, b1, (short)0, acc[1][1], false, false);
        }

        if (hasNext) {
            #pragma unroll
            for (int i = 0; i < 16; ++i) {
                int flat = tid + i * 256;
                int kr = flat >> 7, c = flat & 127;
                ldsB[buf ^ 1][c][kr] = rB[i];
            }
            __builtin_amdgcn_s_wait_tensorcnt(0);
        }
        __syncthreads();
        buf ^= 1;
    }

    // epilogue: C layout = VGPR e -> M = e + 8*half ; N = lane&15
    #pragma unroll
    for (int si = 0; si < 2; ++si)
        #pragma unroll
        for (int sj = 0; sj < 2; ++sj)
            #pragma unroll
            for (int e = 0; e < 8; ++e) {
                int row = rowBase + waveRow * 32 + si * 16 + e + half * 8;
                int col = colBase + waveCol * 32 + sj * 16 + m16;
                size_t idx = (size_t)row * n + col;
                float v = alpha * acc[si][sj][e];
                if (beta != 0.0f) v += beta * Dm[idx];
                C[idx] = v;
            }
}

// ---------------- small kernels ----------------
__global__ void zero_f32(float* p, int n) {
    int i = blockIdx.x * blockDim.x + threadIdx.x;
    if (i < n) p[i] = 0.0f;
}

__global__ void scale_x(const float* __restrict__ X, const float* __restrict__ logls,
                        float* __restrict__ Xs, float* __restrict__ sq) {
    int i = blockIdx.x * blockDim.x + threadIdx.x;
    if (i >= MN) return;
    float acc = 0.0f;
    #pragma unroll
    for (int d = 0; d < DD; ++d) {
        float v = X[i * DD + d] * expf(-logls[d]);
        Xs[i * DD + d] = v;
        acc += v * v;
    }
    sq[i] = acc;
}

__global__ void build_K(const float* __restrict__ Xs, const float* __restrict__ sq,
                        const float* __restrict__ logvar, float* __restrict__ K) {
    int j = blockIdx.x * 16 + threadIdx.x;
    int i = blockIdx.y * 16 + threadIdx.y;
    float dot = 0.0f;
    #pragma unroll
    for (int d = 0; d < DD; ++d) dot += Xs[i * DD + d] * Xs[j * DD + d];
    float dist = fmaxf(sq[i] + sq[j] - 2.0f * dot, 0.0f);
    float v = expf(logvar[0]) * expf(-0.5f * dist);
    if (i == j) v += 1e-5f;
    K[(size_t)i * MN + j] = v;
}

// max over rows of sum(|K[row,:]|)  (K entries >= 0, so int-bit atomicMax is valid)
__global__ __launch_bounds__(256)
void row_abs_max(const float* __restrict__ K, float* normSlot) {
    __shared__ float red[256];
    int row = blockIdx.x;
    float s = 0.0f;
    for (int c = threadIdx.x; c < MN; c += 256) s += fabsf(K[(size_t)row * MN + c]);
    red[threadIdx.x] = s;
    __syncthreads();
    for (int off = 128; off > 0; off >>= 1) {
        if (threadIdx.x < off) red[threadIdx.x] += red[threadIdx.x + off];
        __syncthreads();
    }
    if (threadIdx.x == 0) atomicMax((int*)normSlot, __float_as_int(red[0]));
}

__global__ void ns_init(const float* __restrict__ K, const float* __restrict__ normSlot,
                        float* __restrict__ X0) {
    size_t tot = (size_t)MN * MN;
    float nrm = normSlot[0];
    float sc = 1.0f / (nrm * nrm);   // K symmetric: ||K||_1 == ||K||_inf
    for (size_t idx = (size_t)blockIdx.x * blockDim.x + threadIdx.x; idx < tot;
         idx += (size_t)gridDim.x * blockDim.x)
        X0[idx] = K[idx] * sc;
}

__global__ void diag_extract(const float* __restrict__ A, float* __restrict__ d) {
    int i = blockIdx.x * blockDim.x + threadIdx.x;
    if (i < MN) d[i] = A[(size_t)i * MN + i];
}

__global__ void compute_s(const float* s_raw, float* sSlot) {
    if (threadIdx.x == 0 && blockIdx.x == 0)
        sSlot[0] = fmaxf(expf(s_raw[0]), 1e-6f);
}

// pair scatter: LDS-resident f / grad_ll / W accumulation (ds_add_f32 atomics),
// sparse flush to global
__global__ __launch_bounds__(256)
void pair_accum(const float* __restrict__ f, const int* __restrict__ pw,
                const int* __restrict__ pl, const float* __restrict__ sPtr,
                float* __restrict__ gll, float* __restrict__ Wd, int P) {
    __shared__ float sF[MN];
    __shared__ float sG[MN];
    __shared__ float sW[MN];
    for (int i = threadIdx.x; i < MN; i += 256) {
        sF[i] = f[i];
        sG[i] = 0.0f;
        sW[i] = 0.0f;
    }
    __syncthreads();
    const float s = sPtr[0];
    for (int idx = blockIdx.x * 256 + threadIdx.x; idx < P; idx += gridDim.x * 256) {
        int w = pw[idx], l = pl[idx];
        float diff = sF[w] - sF[l];
        float p = 1.0f / (1.0f + expf(-s * diff));
        p = fminf(fmaxf(p, 1e-7f), 1.0f - 1e-7f);
        float r = s * (1.0f - p);
        float wv = s * s * p * (1.0f - p);
        atomicAdd(&sG[w], r);
        atomicAdd(&sG[l], -r);
        atomicAdd(&sW[w], wv);
        atomicAdd(&sW[l], wv);
    }
    __syncthreads();
    for (int i = threadIdx.x; i < MN; i += 256) {
        if (sG[i] != 0.0f) atomicAdd(&gll[i], sG[i]);
        if (sW[i] != 0.0f) atomicAdd(&Wd[i], sW[i]);
    }
}

// y = A @ x, wave32 per row, x staged in LDS
__global__ __launch_bounds__(256)
void matvec(const float* __restrict__ A, const float* __restrict__ x, float* __restrict__ y) {
    __shared__ float sx[MN];
    for (int i = threadIdx.x; i < MN; i += 256) sx[i] = x[i];
    __syncthreads();
    int wave = threadIdx.x >> 5, lane = threadIdx.x & 31;
    int row = blockIdx.x * 8 + wave;
    float s = 0.0f;
    for (int c = lane; c < MN; c += 32) s += A[(size_t)row * MN + c] * sx[c];
    #pragma unroll
    for (int off = 16; off > 0; off >>= 1) s += __shfl_down(s, off, 32);
    if (lane == 0) y[row] = s;
}

__global__ void newton_update(float* __restrict__ f, const float* __restrict__ gll,
                              const float* __restrict__ Kinvf, const float* __restrict__ Wd,
                              const float* __restrict__ dK) {
    int i = blockIdx.x * blockDim.x + threadIdx.x;
    if (i >= MN) return;
    float w = fmaxf(Wd[i], 1e-8f);
    float grad = gll[i] - Kinvf[i];
    float Hi = -1.0f / (w + dK[i]);
    f[i] += 0.5f * Hi * grad;
}

__global__ __launch_bounds__(256)
void loglik_reduce(const float* __restrict__ f, const int* __restrict__ pw,
                   const int* __restrict__ pl, const float* __restrict__ sPtr,
                   float* accum, int P) {
    __shared__ float sF[MN];
    __shared__ float red[256];
    for (int i = threadIdx.x; i < MN; i += 256) sF[i] = f[i];
    __syncthreads();
    const float s = sPtr[0];
    float sum = 0.0f;
    for (int idx = blockIdx.x * 256 + threadIdx.x; idx < P; idx += gridDim.x * 256) {
        float diff = sF[pw[idx]] - sF[pl[idx]];
        float p = 1.0f / (1.0f + expf(-s * diff));
        sum += logf(p + 1e-12f);
    }
    red[threadIdx.x] = sum;
    __syncthreads();
    for (int off = 128; off > 0; off >>= 1) {
        if (threadIdx.x < off) red[threadIdx.x] += red[threadIdx.x + off];
        __syncthreads();
    }
    if (threadIdx.x == 0) atomicAdd(accum, red[0]);
}

__global__ __launch_bounds__(256)
void dot_reduce(const float* __restrict__ a, const float* __restrict__ b, float* accum) {
    __shared__ float red[256];
    float s = 0.0f;
    for (int i = blockIdx.x * 256 + threadIdx.x; i < MN; i += gridDim.x * 256)
        s += a[i] * b[i];
    red[threadIdx.x] = s;
    __syncthreads();
    for (int off = 128; off > 0; off >>= 1) {
        if (threadIdx.x < off) red[threadIdx.x] += red[threadIdx.x + off];
        __syncthreads();
    }
    if (threadIdx.x == 0) atomicAdd(accum, red[0]);
}

__global__ void finalize(const float* __restrict__ acc /*[loglik, fKf]*/,
                         const float* __restrict__ s_raw, float* __restrict__ out) {
    if (threadIdx.x == 0 && blockIdx.x == 0) {
        float sr = s_raw[0];
        const float sigma = 0.25f;
        float lps = -0.5f * (sr / sigma) * (sr / sigma) - sr - logf(sigma)
                    - 0.91893853320467274178f;  // 0.5*log(2*pi)
        out[0] = acc[0] - 0.5f * acc[1] + lps;
    }
}

// ---------------- host orchestration ----------------
extern "C" void kernel_launch(void* const* d_in, const int* in_sizes, int n_in,
                              void* d_out, int out_size, void* d_ws, size_t ws_size,
                              hipStream_t stream) {
    const float* X_all   = (const float*)d_in[0];
    const int*   pairs_w = (const int*)d_in[1];
    const int*   pairs_l = (const int*)d_in[2];
    const float* log_ls  = (const float*)d_in[3];
    const float* log_var = (const float*)d_in[4];
    const float* s_raw   = (const float*)d_in[5];
    const int    P       = in_sizes[1];
    float* out = (float*)d_out;

    // workspace layout
    char* ws = (char*)d_ws;
    const size_t mm = (size_t)MN * MN * sizeof(float);   // 64 MB
    float* K  = (float*)(ws + 0 * mm);
    float* X0 = (float*)(ws + 1 * mm);
    float* X1 = (float*)(ws + 2 * mm);
    float* T  = (float*)(ws + 3 * mm);
    char* sm  = ws + 4 * mm;
    float* Xs    = (float*)sm;  sm += (size_t)MN * DD * sizeof(float);
    float* sq    = (float*)sm;  sm += (size_t)MN * sizeof(float);
    float* fmap  = (float*)sm;  sm += (size_t)MN * sizeof(float);
    float* gll   = (float*)sm;  sm += (size_t)MN * sizeof(float);   // grad_ll
    float* Wd    = (float*)sm;  sm += (size_t)MN * sizeof(float);   // contiguous after gll
    float* Kinvf = (float*)sm;  sm += (size_t)MN * sizeof(float);
    float* dK    = (float*)sm;  sm += (size_t)MN * sizeof(float);
    float* scal  = (float*)sm;  // [0]=s  [1]=norm(max rowsum)  [2]=loglik  [3]=fKf

    const dim3 gemmGrid(MN / BN, MN / BM);
    const dim3 gemmBlk(256);

    // ---- scalars & K ----
    zero_f32<<<1, 256, 0, stream>>>(scal, 4);
    scale_x<<<(MN + 255) / 256, 256, 0, stream>>>(X_all, log_ls, Xs, sq);
    build_K<<<dim3(MN / 16, MN / 16), dim3(16, 16), 0, stream>>>(Xs, sq, log_var, K);

    // ---- Newton-Schulz inverse: X <- 2X - X(KX), X0 = K / ||K||^2 ----
    row_abs_max<<<MN, 256, 0, stream>>>(K, &scal[1]);
    ns_init<<<2048, 256, 0, stream>>>(K, &scal[1], X0);
    float* cur = X0;
    float* nxt = X1;
    for (int it = 0; it < NS_ITERS; ++it) {
        gemm_wmma_f32<<<gemmGrid, gemmBlk, 0, stream>>>(T, K, cur, T, 1.0f, 0.0f, MN);
        gemm_wmma_f32<<<gemmGrid, gemmBlk, 0, stream>>>(nxt, cur, T, cur, -1.0f, 2.0f, MN);
        float* tmp = cur; cur = nxt; nxt = tmp;
    }
    float* Kinv = cur;
    diag_extract<<<(MN + 255) / 256, 256, 0, stream>>>(Kinv, dK);

    // ---- Newton iterations for f_map ----
    zero_f32<<<(MN + 255) / 256, 256, 0, stream>>>(fmap, MN);
    compute_s<<<1, 32, 0, stream>>>(s_raw, &scal[0]);
    for (int it = 0; it < NEWTON_ITERS; ++it) {
        zero_f32<<<(2 * MN + 255) / 256, 256, 0, stream>>>(gll, 2 * MN);  // gll + Wd
        pair_accum<<<512, 256, 0, stream>>>(fmap, pairs_w, pairs_l, &scal[0], gll, Wd, P);
        matvec<<<MN / 8, 256, 0, stream>>>(Kinv, fmap, Kinvf);
        newton_update<<<(MN + 255) / 256, 256, 0, stream>>>(fmap, gll, Kinvf, Wd, dK);
    }

    // ---- posterior value ----
    matvec<<<MN / 8, 256, 0, stream>>>(Kinv, fmap, Kinvf);
    loglik_reduce<<<1024, 256, 0, stream>>>(fmap, pairs_w, pairs_l, &scal[0], &scal[2], P);
    dot_reduce<<<16, 256, 0, stream>>>(fmap, Kinvf, &scal[3]);
    finalize<<<1, 32, 0, stream>>>(&scal[2], s_raw, out);
    (void)n_in; (void)out_size; (void)ws_size;
}